// AttentionPooling1D_17540646437366
// MI455X (gfx1250) — compile-verified
//
#include <hip/hip_runtime.h>

// ---------------------------------------------------------------------------
// AttentionPooling1D, one-pass split-softmax for MI455X (gfx1250)
//   x: [B=64, S=4096, D=1024] f32   mask: [B,S] i32   w: [1,D] f32
//   out: [B, D] f32
// Pass 1: per (b, 32-row chunk) block:
//   - TDM tensor_load_to_lds DMAs the 128 KB x tile into LDS (TENSORcnt)
//   - scores via v_wmma_f32_16x16x4_f32, parallelized over all 8 waves
//     (2 row-groups x 4 D-segments, partials reduced through LDS)
//   - masked local softmax stats (m_c, l_c) + local weighted acc_c[D]
// Pass 2: per-batch merge of 128 chunks with exp(m_c - M)/L rescale.
// ---------------------------------------------------------------------------

#define B_      64
#define S_      4096
#define D_      1024
#define SC      32                 // s-rows per chunk (tile = 128 KB in LDS)
#define NCHUNK  (S_ / SC)          // 128
#define THREADS 256                // 8 wave32
#define NSEG    4                  // D-segments for the WMMA score phase
#define DSEG    (D_ / NSEG)        // 256

typedef __attribute__((ext_vector_type(2))) float        v2f;
typedef __attribute__((ext_vector_type(8))) float        v8f;
typedef __attribute__((ext_vector_type(4))) unsigned int u32x4;
typedef __attribute__((ext_vector_type(8))) int          i32x8;
typedef __attribute__((ext_vector_type(4))) int          i32x4;

__device__ __forceinline__ unsigned int lds_off(const void* p) {
  // generic->LDS aperture: low 32 bits carry the LDS byte offset
  return (unsigned int)(unsigned long long)p;
}

extern "C" __global__ __launch_bounds__(THREADS)
void ap1d_partial(const float* __restrict__ x,
                  const int*   __restrict__ mask,
                  const float* __restrict__ w,
                  float* __restrict__ ws_m,
                  float* __restrict__ ws_l,
                  float* __restrict__ ws_acc)
{
  extern __shared__ unsigned char smem_raw[];
  float* shX    = (float*)smem_raw;       // SC * D_ floats (128 KB)
  float* shW    = shX + SC * D_;          // D_ floats (w staged in LDS)
  float* shPart = shW + D_;               // NSEG * SC partial scores
  float* shS    = shPart + NSEG * SC;     // SC masked scores
  float* shP    = shS + SC;               // SC probabilities (unnormalized)

  const int blk   = blockIdx.x;
  const int b     = blk / NCHUNK;
  const int chunk = blk % NCHUNK;
  const int s0    = chunk * SC;
  const int tid   = threadIdx.x;
  const int wave  = tid >> 5;
  const int lane  = tid & 31;

  // -------- Phase 0: TDM bulk DMA of x[b, s0:s0+SC, :] into LDS ----------
  if (wave == 0) {
    const unsigned long long gaddr =
        (unsigned long long)(const void*)(x + ((size_t)b * S_ + (size_t)s0) * D_);
    u32x4 g0;
    g0.x = 1u;                                             // count=1, user mode
    g0.y = lds_off(shX);                                   // lds_addr (bytes)
    g0.z = (unsigned int)(gaddr & 0xFFFFFFFFull);          // global_addr[31:0]
    g0.w = (unsigned int)((gaddr >> 32) & 0x1FFFFFFull)    // global_addr[56:32]
         | (2u << 30);                                     // type = 2 ("image")
    i32x8 g1;
    g1[0] = (int)(2u << 16);                 // data_size = 4 bytes
    g1[1] = (int)((unsigned)D_ << 16);       // tensor_dim0 = 1024 (lo16 @ b63:48)
    g1[2] = (int)((unsigned)SC << 16);       // tensor_dim1 = 32  (lo16 @ b95:80)
    g1[3] = (int)((unsigned)D_ << 16);       // tile_dim0 = 1024  (b127:112)
    g1[4] = (int)SC;                         // tile_dim1 = 32, tile_dim2 = 0
    g1[5] = (int)D_;                         // tensor_dim0_stride = 1024
    g1[6] = 0;                               // stride hi / dim1_stride lo
    g1[7] = 0;                               // dim1_stride hi
    i32x4 gz = {0, 0, 0, 0};
#if defined(__clang_major__) && (__clang_major__ >= 23)
    i32x8 gz8 = {0, 0, 0, 0, 0, 0, 0, 0};
    __builtin_amdgcn_tensor_load_to_lds(g0, g1, gz, gz, gz8, 0);
#else
    __builtin_amdgcn_tensor_load_to_lds(g0, g1, gz, gz, 0);
#endif
  }

  // stage w into LDS while the TDM transfer is in flight
  for (int i = tid; i < D_; i += THREADS) shW[i] = w[i];

  if (wave == 0) __builtin_amdgcn_s_wait_tensorcnt(0);
  __syncthreads();

  // -------- Phase 1: scores[s] = dot(x[s,:], w) via fp32 WMMA ------------
  // All 8 waves participate: wave = rowgrp (2) x dseg (4). Each wave runs a
  // 64-long v_wmma_f32_16x16x4_f32 accumulation chain over its 256-wide D
  // segment; the 4 segment partials per row are reduced through LDS.
  // B (= w) is broadcast across the 16 output columns: column-redundant but
  // free — this phase rides the matrix pipe while the kernel is HBM-bound.
  {
    const int rowgrp = wave & 1;
    const int dseg   = wave >> 1;
    const int mrow   = (rowgrp << 4) + (lane & 15);  // A row fed by this lane
    const int khalf  = (lane >> 4) * 2;  // lanes 0-15: K=0,1; 16-31: K=2,3
    const float* xrow = shX + mrow * D_;
    v8f c = {0.f, 0.f, 0.f, 0.f, 0.f, 0.f, 0.f, 0.f};
    const int d_lo = dseg * DSEG, d_hi = d_lo + DSEG;
    for (int d0 = d_lo; d0 < d_hi; d0 += 4) {
      v2f a, bb;
      a.x  = xrow[d0 + khalf];
      a.y  = xrow[d0 + khalf + 1];
      bb.x = shW[d0 + khalf];
      bb.y = shW[d0 + khalf + 1];
      c = __builtin_amdgcn_wmma_f32_16x16x4_f32(false, a, false, bb,
                                                (short)0, c, false, false);
    }
    // C/D layout: lanes 0-15 col n, VGPR r = row r; lanes 16-31 = row r+8.
    // Column 0 of each half (lane 0 / lane 16) carries the 16 row sums.
    if ((lane & 15) == 0) {
      const int rbase = (rowgrp << 4) + ((lane >> 4) ? 8 : 0);
#pragma unroll
      for (int r = 0; r < 8; ++r) shPart[dseg * SC + rbase + r] = c[r];
    }
  }
  __syncthreads();

  // -------- Phase 2: masked chunk-local softmax stats --------------------
  if (tid < SC) {
    const float sc = shPart[0 * SC + tid] + shPart[1 * SC + tid]
                   + shPart[2 * SC + tid] + shPart[3 * SC + tid];
    const int   mk = mask[(size_t)b * S_ + (size_t)(s0 + tid)];
    shS[tid] = mk ? sc : -__builtin_inff();
  }
  __syncthreads();
  if (tid == 0) {
    float m = -__builtin_inff();
    for (int i = 0; i < SC; ++i) m = fmaxf(m, shS[i]);
    float l = 0.f;
    if (m > -__builtin_inff()) {
      for (int i = 0; i < SC; ++i) {
        const float p = __expf(shS[i] - m);   // exp(-inf - m) == 0 for masked
        shP[i] = p;
        l += p;
      }
    } else {
      for (int i = 0; i < SC; ++i) shP[i] = 0.f;  // fully-masked chunk
    }
    ws_m[blk] = m;
    ws_l[blk] = l;
  }
  __syncthreads();

  // -------- Phase 3: chunk-local weighted sum acc[d] = sum_s p[s]*x[s,d] --
  {
    const int d = tid * 4;                       // 256 threads x 4 = 1024
    float4 acc = {0.f, 0.f, 0.f, 0.f};
    for (int s = 0; s < SC; ++s) {
      const float  p  = shP[s];
      const float4 xv = *(const float4*)(shX + s * D_ + d);
      acc.x += p * xv.x; acc.y += p * xv.y;
      acc.z += p * xv.z; acc.w += p * xv.w;
    }
    *(float4*)(ws_acc + (size_t)blk * D_ + d) = acc;
  }
}

extern "C" __global__ __launch_bounds__(THREADS)
void ap1d_finalize(const float* __restrict__ ws_m,
                   const float* __restrict__ ws_l,
                   const float* __restrict__ ws_acc,
                   float* __restrict__ out)
{
  __shared__ float sh_m[NCHUNK];
  __shared__ float sh_l[NCHUNK];
  __shared__ float sh_scale[NCHUNK];
  const int b   = blockIdx.x;
  const int tid = threadIdx.x;

  for (int c = tid; c < NCHUNK; c += THREADS) {
    sh_m[c] = ws_m[b * NCHUNK + c];
    sh_l[c] = ws_l[b * NCHUNK + c];
  }
  __syncthreads();
  if (tid == 0) {
    float M = -__builtin_inff();
    for (int c = 0; c < NCHUNK; ++c) M = fmaxf(M, sh_m[c]);
    float L = 0.f;
    for (int c = 0; c < NCHUNK; ++c) {
      const float e = (sh_m[c] > -__builtin_inff()) ? __expf(sh_m[c] - M) : 0.f;
      sh_scale[c] = e;
      L += e * sh_l[c];
    }
    const float inv = (L > 0.f) ? 1.f / L : 0.f;
    for (int c = 0; c < NCHUNK; ++c) sh_scale[c] *= inv;
  }
  __syncthreads();

  const int d = tid * 4;
  float4 acc = {0.f, 0.f, 0.f, 0.f};
  for (int c = 0; c < NCHUNK; ++c) {
    const float  s = sh_scale[c];
    const float4 v = *(const float4*)(ws_acc + ((size_t)b * NCHUNK + c) * D_ + d);
    acc.x += s * v.x; acc.y += s * v.y; acc.z += s * v.z; acc.w += s * v.w;
  }
  *(float4*)(out + (size_t)b * D_ + d) = acc;
}

extern "C" void kernel_launch(void* const* d_in, const int* in_sizes, int n_in,
                              void* d_out, int out_size, void* d_ws, size_t ws_size,
                              hipStream_t stream) {
  (void)in_sizes; (void)n_in; (void)out_size; (void)ws_size;
  const float* x    = (const float*)d_in[0];
  const int*   mask = (const int*)d_in[1];
  const float* w    = (const float*)d_in[2];
  float*       out  = (float*)d_out;

  float* wsf    = (float*)d_ws;
  float* ws_m   = wsf;                                  // [B*NCHUNK]
  float* ws_l   = wsf + (size_t)B_ * NCHUNK;            // [B*NCHUNK]
  float* ws_acc = wsf + 2 * (size_t)B_ * NCHUNK;        // [B*NCHUNK, D]

  const size_t shbytes =
      (size_t)(SC * D_ + D_ + NSEG * SC + 2 * SC) * sizeof(float); // ~136 KB
  ap1d_partial<<<dim3(B_ * NCHUNK), dim3(THREADS), shbytes, stream>>>(
      x, mask, w, ws_m, ws_l, ws_acc);
  ap1d_finalize<<<dim3(B_), dim3(THREADS), 0, stream>>>(ws_m, ws_l, ws_acc, out);
}